// SSIMLoss_75179107549981
// MI455X (gfx1250) — compile-verified
//
#include <hip/hip_runtime.h>
#include <hip/hip_bf16.h>
#include <stdint.h>

// ---------------------------------------------------------------------------
// SSIM (skimage, win=7, valid region) fused kernel for gfx1250 (MI455X).
//   - TDM (tensor_load_to_lds) stages 80x82 fp32 tiles of both images to LDS
//   - 7x7 box sums computed as banded GEMMs on V_WMMA_F32_16X16X4_F32
//   - constants always on one WMMA operand; D->B operand relayout done with
//     lane shuffles (self / lane^16 pattern), no LDS transpose
//   - fp64 atomic accumulation, finalize kernel emits the mean
// ---------------------------------------------------------------------------

typedef float v2f __attribute__((ext_vector_type(2)));
typedef float v8f __attribute__((ext_vector_type(8)));
typedef unsigned int v4u __attribute__((ext_vector_type(4)));
typedef int v4i __attribute__((ext_vector_type(4)));
typedef int v8i __attribute__((ext_vector_type(8)));

#define IMG   4096
#define OUTN  4090         // 4096 - 7 + 1
#define ROWS  80           // LDS tile rows   (64 out + 6 halo + band slack)
#define PITCH 82           // LDS tile pitch  (even -> b64-aligned A loads)

static __device__ __forceinline__ v8f wmma4(v2f a, v2f b, v8f c) {
  // D = A(16x4,f32) * B(4x16,f32) + C(16x16,f32)
  return __builtin_amdgcn_wmma_f32_16x16x4_f32(false, a, false, b, (short)0, c,
                                               false, false);
}

static __device__ __forceinline__ float bandv(int a, int k) {
  // 7-wide band: 1 iff a <= k <= a+6
  return (k >= a && k <= a + 6) ? 1.0f : 0.0f;
}

static __device__ __forceinline__ float chv(float o, float t, int ch) {
  if (ch == 0) return o;
  if (ch == 1) return t;
  if (ch == 2) return o * o;
  if (ch == 3) return t * t;
  return o * t;
}

// Issue one TDM descriptor: load a ROWSxPITCH fp32 tile whose top-left element
// is (Y0, X0) of a 4096x4096 row-major tensor into LDS at byte offset ldsOff.
// tensor_dim is clamped to the image so out-of-image reads return zeros.
static __device__ __forceinline__ void tdm_load_tile(unsigned ldsOff,
                                                     const float* gbase,
                                                     unsigned X0, unsigned Y0) {
  unsigned long long ga = (unsigned long long)(uintptr_t)gbase +
                          ((unsigned long long)Y0 * IMG + X0) * 4ull;
  unsigned td0 = IMG - X0;  // remaining valid columns from tile start
  unsigned td1 = IMG - Y0;  // remaining valid rows
  v4u g0;
  g0.x = 1u;                                   // count=1 (valid descriptor)
  g0.y = ldsOff;                               // lds_addr
  g0.z = (unsigned)ga;                         // global_addr[31:0]
  g0.w = (unsigned)((ga >> 32) & 0x1FFFFFFull) // global_addr[56:32]
         | (2u << 30);                         // type = 2 ("image")
  v8i g1;
  g1[0] = 0x20000;                                        // data_size=2 (4B)
  g1[1] = (int)((td0 & 0xFFFFu) << 16);                   // tensor_dim0 lo16
  g1[2] = (int)((td0 >> 16) | ((td1 & 0xFFFFu) << 16));   // td0 hi / td1 lo
  g1[3] = (int)((td1 >> 16) | ((unsigned)PITCH << 16));   // td1 hi / tile_dim0
  g1[4] = ROWS;                                           // tile_dim1
  g1[5] = IMG;                                            // dim0_stride lo32
  g1[6] = 0;
  g1[7] = 0;
  v4i z4 = {0, 0, 0, 0};               // groups 2/3 unused (2-D tensor)
  v8i z8 = {0, 0, 0, 0, 0, 0, 0, 0};   // extra group in 6-arg builtin form
  __builtin_amdgcn_tensor_load_to_lds(g0, g1, z4, z4, z8, 0);
}

__global__ void ssim_zero_kernel(double* acc) { acc[0] = 0.0; }

__global__ void ssim_finalize_kernel(const double* acc, float* out) {
  out[0] = (float)(acc[0] / ((double)OUTN * (double)OUTN));
}

__global__ __launch_bounds__(512) void ssim_main_kernel(
    const float* __restrict__ O, const float* __restrict__ T,
    double* __restrict__ acc) {
  __shared__ float tO[ROWS * PITCH];
  __shared__ float tT[ROWS * PITCH];
  __shared__ float red[16];

  const int X0 = blockIdx.x * 64;   // output/input tile origin (window start)
  const int Y0 = blockIdx.y * 64;

  const int wave = threadIdx.x >> 5;
  const int lane = threadIdx.x & 31;
  const int ln15 = lane & 15;
  const int hi   = lane >> 4;

  // ---- stage tiles via Tensor Data Mover (wave0 -> O, wave1 -> T) ---------
  if (wave == 0) {
    tdm_load_tile((unsigned)(uintptr_t)(void*)tO, O, (unsigned)X0, (unsigned)Y0);
    __builtin_amdgcn_s_wait_tensorcnt(0);
  } else if (wave == 1) {
    tdm_load_tile((unsigned)(uintptr_t)(void*)tT, T, (unsigned)X0, (unsigned)Y0);
    __builtin_amdgcn_s_wait_tensorcnt(0);
  }
  __syncthreads();

  // ---- per-wave 16x16 output sub-tile -------------------------------------
  const int r0  = (wave >> 2) * 16;   // sub-tile row origin within WG tile
  const int c0w = (wave & 3) * 16;    // sub-tile col origin within WG tile

  // Band constants (shared by stage-1 B operand and stage-2 A operand):
  // cK1[kk] = band(ln15, 4kk+2hi + {0,1}) ; cK2 = same with K offset 16.
  v2f cK1[4], cK2[4];
#pragma unroll
  for (int kk = 0; kk < 4; ++kk) {
    int k0 = 4 * kk + 2 * hi;
    cK1[kk].x = bandv(ln15, k0);      cK1[kk].y = bandv(ln15, k0 + 1);
    cK2[kk].x = bandv(ln15, k0 + 16); cK2[kk].y = bandv(ln15, k0 + 17);
  }

  // A-layout tile loads: element (m,k) = X[r0+16*rbs+m][c0w+16*cs+k]
  float aO[2][2][4][2], aT[2][2][4][2];
#pragma unroll
  for (int rbs = 0; rbs < 2; ++rbs) {
    int row = r0 + 16 * rbs + ln15;
#pragma unroll
    for (int cs = 0; cs < 2; ++cs) {
      int cbase = c0w + 16 * cs + 2 * hi;
#pragma unroll
      for (int kk = 0; kk < 4; ++kk) {
#pragma unroll
        for (int j = 0; j < 2; ++j) {
          int col = cbase + 4 * kk + j;
          aO[rbs][cs][kk][j] = tO[row * PITCH + col];
          aT[rbs][cs][kk][j] = tT[row * PITCH + col];
        }
      }
    }
  }

  // ---- 5 channels: box sums via banded WMMA GEMMs -------------------------
  v8f Sacc[5];
#pragma unroll
  for (int ch = 0; ch < 5; ++ch) {
    // Stage 1: W_top/W_bot = X_rows(16x32pad) * B_band  (horizontal 7-sums)
    v8f wtb[2];
#pragma unroll
    for (int rbs = 0; rbs < 2; ++rbs) {
      v8f w = {0.f, 0.f, 0.f, 0.f, 0.f, 0.f, 0.f, 0.f};
#pragma unroll
      for (int cs = 0; cs < 2; ++cs) {
#pragma unroll
        for (int kk = 0; kk < 4; ++kk) {
          v2f a;
          a.x = chv(aO[rbs][cs][kk][0], aT[rbs][cs][kk][0], ch);
          a.y = chv(aO[rbs][cs][kk][1], aT[rbs][cs][kk][1], ch);
          w = wmma4(a, (cs ? cK2[kk] : cK1[kk]), w);
        }
      }
      wtb[rbs] = w;
    }

    // Stage 2: Out = A_band * W  (vertical 7-sums). B operand W[k][n] comes
    // from the D-layout of wtb via the self/lane^16 shuffle pattern.
    v8f out = {0.f, 0.f, 0.f, 0.f, 0.f, 0.f, 0.f, 0.f};
#pragma unroll
    for (int rb = 0; rb < 2; ++rb) {
#pragma unroll
      for (int kk = 0; kk < 4; ++kk) {
        v2f b;
#pragma unroll
        for (int j = 0; j < 2; ++j) {
          float val;
          if (kk < 2) {
            float t = __shfl(wtb[rb][4 * kk + 2 + j], ln15, 32);
            val = hi ? t : wtb[rb][4 * kk + j];
          } else {
            float t = __shfl(wtb[rb][4 * (kk - 2) + j], ln15 + 16, 32);
            val = hi ? wtb[rb][4 * (kk - 2) + 2 + j] : t;
          }
          if (j == 0) b.x = val; else b.y = val;
        }
        out = wmma4((rb ? cK2[kk] : cK1[kk]), b, out);
      }
    }
    Sacc[ch] = out;
  }

  // ---- SSIM formula + masked accumulation ---------------------------------
  const float inv49 = 1.0f / 49.0f;
  const float cov   = 49.0f / 48.0f;
  const float C1c   = 0.0004f;   // (0.01*2)^2
  const float C2c   = 0.0036f;   // (0.03*2)^2
  float part = 0.0f;
#pragma unroll
  for (int e = 0; e < 8; ++e) {
    int M  = e + 8 * hi;
    int gy = Y0 + r0 + M;
    int gx = X0 + c0w + ln15;
    float ux  = Sacc[0][e] * inv49;
    float uy  = Sacc[1][e] * inv49;
    float uxx = Sacc[2][e] * inv49;
    float uyy = Sacc[3][e] * inv49;
    float uxy = Sacc[4][e] * inv49;
    float vx  = cov * (uxx - ux * ux);
    float vy  = cov * (uyy - uy * uy);
    float vxy = cov * (uxy - ux * uy);
    float A1v = 2.0f * ux * uy + C1c;
    float A2v = 2.0f * vxy + C2c;
    float B1v = ux * ux + uy * uy + C1c;
    float B2v = vx + vy + C2c;
    float S   = (A1v * A2v) / (B1v * B2v);
    part += (gy < OUTN && gx < OUTN) ? S : 0.0f;
  }

  // ---- reduce: wave -> block -> global fp64 atomic ------------------------
#pragma unroll
  for (int off = 16; off > 0; off >>= 1) part += __shfl_down(part, off, 32);
  if (lane == 0) red[wave] = part;
  __syncthreads();
  if (threadIdx.x == 0) {
    float t = 0.0f;
#pragma unroll
    for (int i = 0; i < 16; ++i) t += red[i];
    atomicAdd(acc, (double)t);
  }
}

extern "C" void kernel_launch(void* const* d_in, const int* in_sizes, int n_in,
                              void* d_out, int out_size, void* d_ws,
                              size_t ws_size, hipStream_t stream) {
  (void)in_sizes; (void)n_in; (void)out_size; (void)ws_size;
  const float* O = (const float*)d_in[0];
  const float* T = (const float*)d_in[1];
  double* acc = (double*)d_ws;
  float* out = (float*)d_out;

  ssim_zero_kernel<<<dim3(1), dim3(1), 0, stream>>>(acc);
  dim3 grid(64, 64);  // 64 tiles of 64x64 outputs cover 4090x4090
  ssim_main_kernel<<<grid, dim3(512), 0, stream>>>(O, T, acc);
  ssim_finalize_kernel<<<dim3(1), dim3(1), 0, stream>>>(acc, out);
}